// GHHeatmapLoss_52561809768998
// MI455X (gfx1250) — compile-verified
//
#include <hip/hip_runtime.h>
#include <hip/hip_bf16.h>
#include <stdint.h>

#define GHM_BINS   10
#define GHM_GRID   1024
#define GHM_BLOCK  256
#define GHM_WAVES  (GHM_BLOCK / 32)

// ---------------------------------------------------------------------------
// CDNA5 async global->LDS copy (ASYNCcnt-tracked), 16 bytes per lane.
// GVS addressing: scalar 64-bit base plus per-lane 32-bit byte offset.
// ---------------------------------------------------------------------------
__device__ __forceinline__ void async_ld16(uint32_t lds_addr, uint64_t base,
                                           uint32_t byte_off) {
    asm volatile("global_load_async_to_lds_b128 %0, %1, %2"
                 :: "v"(lds_addr), "v"(byte_off), "s"(base) : "memory");
}
#define WAIT_ASYNC(N) asm volatile("s_wait_asynccnt " #N ::: "memory")

__device__ __forceinline__ uint32_t lds_off(const void* p) {
    return (uint32_t)(uintptr_t)p;   // low 32 bits of generic addr == LDS byte offset
}

// ---------------------------------------------------------------------------
// Per-element GHM work: bin + log term into 10 register accumulators + mask.
// ---------------------------------------------------------------------------
__device__ __forceinline__ void ghm_elem(float pr, float gv,
                                         float acc[GHM_BINS], unsigned& mask) {
    float diff = fabsf(pr - gv);
    int b = (int)(diff * 10.0f);
    b = b > 9 ? 9 : b;
    float x = (gv == 1.0f) ? pr : (1.0f - pr);   // log(pre) or log1p(-pre)
    float t = __logf(x);
    mask |= (1u << b);
#pragma unroll
    for (int k = 0; k < GHM_BINS; ++k)
        acc[k] += (b == k) ? t : 0.0f;
}

// ---------------------------------------------------------------------------
// Pass 1: stream pre/gt via async-to-LDS double buffer, accumulate per-bin
// log-term sums + nonempty mask; fixed-order block reduction -> partials.
// partials layout: [GHM_GRID][16] floats (bins 0..9 valid, 10..15 zero).
// ---------------------------------------------------------------------------
__global__ __launch_bounds__(GHM_BLOCK)
void ghm_pass1(const float* __restrict__ pre, const float* __restrict__ gt,
               float* __restrict__ partials, unsigned* __restrict__ masks,
               long n) {
    __shared__ float4 sh_pre[2][GHM_BLOCK];
    __shared__ float4 sh_gt[2][GHM_BLOCK];
    __shared__ float  sh_red[GHM_WAVES][GHM_BINS];
    __shared__ unsigned sh_m[GHM_WAVES];

    const int  tid      = threadIdx.x;
    const long tglob    = (long)blockIdx.x * GHM_BLOCK + tid;
    const long nthreads = (long)gridDim.x * GHM_BLOCK;
    const long n4       = n >> 2;
    const long iters    = n4 / nthreads;          // uniform across all threads

    float acc[GHM_BINS];
#pragma unroll
    for (int k = 0; k < GHM_BINS; ++k) acc[k] = 0.0f;
    unsigned mask = 0;

    const uint64_t pbase = (uint64_t)(uintptr_t)pre;
    const uint64_t gbase = (uint64_t)(uintptr_t)gt;
    const uint32_t lp0 = lds_off(&sh_pre[0][tid]);
    const uint32_t lp1 = lds_off(&sh_pre[1][tid]);
    const uint32_t lg0 = lds_off(&sh_gt[0][tid]);
    const uint32_t lg1 = lds_off(&sh_gt[1][tid]);

    if (iters > 0) {
        uint32_t off = (uint32_t)(tglob * 16);          // float4 byte offset
        const uint32_t stride = (uint32_t)(nthreads * 16);
        // prologue: fill stage 0
        async_ld16(lp0, pbase, off);
        async_ld16(lg0, gbase, off);
        int stage = 0;
        for (long it = 0; it < iters; ++it) {
            if (it + 1 < iters) {
                uint32_t noff = off + stride;
                async_ld16(stage ? lp0 : lp1, pbase, noff);
                async_ld16(stage ? lg0 : lg1, gbase, noff);
                WAIT_ASYNC(2);      // current stage's 2 loads complete
            } else {
                WAIT_ASYNC(0);
            }
            float4 p = sh_pre[stage][tid];
            float4 g = sh_gt[stage][tid];
            ghm_elem(p.x, g.x, acc, mask);
            ghm_elem(p.y, g.y, acc, mask);
            ghm_elem(p.z, g.z, acc, mask);
            ghm_elem(p.w, g.w, acc, mask);
            off += stride;
            stage ^= 1;
        }
    }
    // scalar tail (covers n%4 and any remainder of n4)
    for (long e = iters * nthreads * 4 + tglob; e < n; e += nthreads)
        ghm_elem(pre[e], gt[e], acc, mask);

    // ---- wave32 reduction (deterministic shuffle tree) ----
#pragma unroll
    for (int k = 0; k < GHM_BINS; ++k) {
        float v = acc[k];
#pragma unroll
        for (int off = 16; off > 0; off >>= 1) v += __shfl_xor(v, off, 32);
        acc[k] = v;
    }
    unsigned m = mask;
#pragma unroll
    for (int off = 16; off > 0; off >>= 1) m |= __shfl_xor(m, off, 32);

    const int wv = tid >> 5, ln = tid & 31;
    if (ln == 0) {
#pragma unroll
        for (int k = 0; k < GHM_BINS; ++k) sh_red[wv][k] = acc[k];
        sh_m[wv] = m;
    }
    __syncthreads();

    if (tid < 16) {
        float s = 0.0f;
        if (tid < GHM_BINS)
            for (int w = 0; w < GHM_WAVES; ++w) s += sh_red[w][tid];  // fixed order
        partials[(long)blockIdx.x * 16 + tid] = s;
    }
    if (tid == 0) {
        unsigned mm = 0;
        for (int w = 0; w < GHM_WAVES; ++w) mm |= sh_m[w];
        masks[blockIdx.x] = mm;
    }
}

// ---------------------------------------------------------------------------
// Pass 2 (single wave32): WMMA column-sum of partials[GHM_GRID][16] with
// A = ones(16x4)  ->  D[m][n] = sum_k B[k][n], then combine with acc_sum.
// out = sum_b nonempty[b] * S[b] / (acc_sum[b] * n)   (tot cancels)
// ---------------------------------------------------------------------------
typedef __attribute__((ext_vector_type(2))) float v2f;
typedef __attribute__((ext_vector_type(8))) float v8f;

__global__ __launch_bounds__(32)
void ghm_pass2(const float* __restrict__ partials,
               const unsigned* __restrict__ masks,
               const float* __restrict__ acc_sum,
               float* __restrict__ out, int nblocks) {
    const int lane = threadIdx.x;

    // OR-reduce nonempty masks (integer: order-independent, deterministic)
    unsigned m = 0;
    for (int i = lane; i < nblocks; i += 32) m |= masks[i];
#pragma unroll
    for (int off = 16; off > 0; off >>= 1) m |= __shfl_xor(m, off, 32);
    const int nbins = __popc(m & 0x3FFu);

    // WMMA reduction: each step folds 4 rows of 16 columns into C.
    v2f a; a.x = 1.0f; a.y = 1.0f;          // ones matrix (K-permutation irrelevant)
    v8f c = {};
    const int half = lane >> 4, col = lane & 15;
    for (int r = 0; r < nblocks; r += 4) {
        v2f b;
        b.x = partials[(r + half) * 16 + col];        // rows r, r+1
        b.y = partials[(r + 2 + half) * 16 + col];    // rows r+2, r+3
        c = __builtin_amdgcn_wmma_f32_16x16x4_f32(
                /*neg_a=*/false, a, /*neg_b=*/false, b,
                /*c_mod=*/(short)0, c, /*reuse_a=*/false, /*reuse_b=*/false);
    }
    // D row M=0 lives in VGPR0, lanes 0-15 (N = lane); all rows equal.
    const float S = c[0];

    float contrib = 0.0f;
    if (lane < GHM_BINS && ((m >> lane) & 1u)) {
        const int nn = nbins > 0 ? nbins : 1;
        contrib = S / (acc_sum[lane] * (float)nn);
    }
#pragma unroll
    for (int off = 16; off > 0; off >>= 1) contrib += __shfl_xor(contrib, off, 32);
    if (lane == 0) out[0] = contrib;
}

// ---------------------------------------------------------------------------
extern "C" void kernel_launch(void* const* d_in, const int* in_sizes, int n_in,
                              void* d_out, int out_size, void* d_ws, size_t ws_size,
                              hipStream_t stream) {
    const float* pre     = (const float*)d_in[0];
    const float* gt      = (const float*)d_in[1];
    const float* acc_sum = (const float*)d_in[2];
    const long   n       = (long)in_sizes[0];

    float*    partials = (float*)d_ws;                                   // GHM_GRID*16 f32
    unsigned* masks    = (unsigned*)((char*)d_ws +
                                     (size_t)GHM_GRID * 16 * sizeof(float)); // GHM_GRID u32

    ghm_pass1<<<GHM_GRID, GHM_BLOCK, 0, stream>>>(pre, gt, partials, masks, n);
    ghm_pass2<<<1, 32, 0, stream>>>(partials, masks, acc_sum, (float*)d_out, GHM_GRID);
}